// TrajectoryAutoregressiveModel_46196668235901
// MI455X (gfx1250) — compile-verified
//
#include <hip/hip_runtime.h>
#include <hip/hip_bf16.h>
#include <math.h>

typedef float v2f __attribute__((ext_vector_type(2)));
typedef float v4f __attribute__((ext_vector_type(4)));
typedef float v8f __attribute__((ext_vector_type(8)));

#define Dm   512
#define Bv   128
#define Sv   30
#define Avn  7
#define Hh   8
#define Lw   6
#define Fv   2048
#define NSEG 100
#define NDET 100
#define NVIS 1024
#define NPT  256
#define NMEM (NSEG + NDET + NVIS + NPT)   // 1480

#define KC   64          // K-chunk staged in LDS
#define SA   68          // padded LDS row stride (68 % 64 == 4 -> bank-conflict-free b64 frag reads)

// ---------------------------------------------------------------------------
// fp32 WMMA GEMM with LDS staging: C = A[M,K] @ W[N,K]^T + bias (+bias2) [ReLU]
// Output row remap: outRow(r) = (r / rowsPerB) * outStrideB + outOff + (r % rowsPerB)
// Requires M % 128 == 0 (true for every call below). N, K may be ragged.
// Block = 128 threads (4 waves). Per K-chunk of 64: block stages A(128x64) and
// W(64x64) into LDS with coalesced b128 loads, then each wave computes a
// 32(M) x 64(N) tile as 2x4 WMMA 16x16 accumulators (V_WMMA_F32_16X16X4_F32),
// reading fragments from LDS via ds_load_b64.
// ---------------------------------------------------------------------------
__global__ __launch_bounds__(128)
void gemm_wmma_f32(const float* __restrict__ A, const float* __restrict__ W,
                   const float* __restrict__ bias, const float* __restrict__ bias2,
                   float* __restrict__ C, int M, int N, int K,
                   int rowsPerB, int outStrideB, int outOff, int relu)
{
    __shared__ float As[128 * SA];
    __shared__ float Ws[64 * SA];

    const int tid  = threadIdx.x;
    const int lane = tid & 31;
    const int wave = tid >> 5;
    const int half = lane >> 4;      // 0: lanes 0-15, 1: lanes 16-31
    const int l16  = lane & 15;
    const int m0b  = blockIdx.y * 128;       // block's first A row
    const int m0   = m0b + wave * 32;        // wave's first A row
    const int n0   = blockIdx.x * 64;        // block's first W row / C col

    v8f acc[2][4];
#pragma unroll
    for (int t = 0; t < 2; ++t)
#pragma unroll
        for (int u = 0; u < 4; ++u) {
            v8f z = {0.f, 0.f, 0.f, 0.f, 0.f, 0.f, 0.f, 0.f};
            acc[t][u] = z;
        }

    const bool k_aligned = ((K & 3) == 0);

    for (int kb = 0; kb < K; kb += KC) {
        // ---- stage A chunk: rows m0b..m0b+127, cols kb..kb+KC ----
#pragma unroll
        for (int it = 0; it < (128 * KC) / (128 * 4); ++it) {  // 16 iters
            const int f   = tid * 4 + it * 512;
            const int row = f >> 6;          // / KC
            const int col = f & (KC - 1);
            const float* src = A + (size_t)(m0b + row) * K + kb + col;
            v4f val;
            if (k_aligned && (kb + col + 4 <= K)) {
                val = *(const v4f*)src;
            } else {
                val.x = (kb + col + 0 < K) ? src[0] : 0.f;
                val.y = (kb + col + 1 < K) ? src[1] : 0.f;
                val.z = (kb + col + 2 < K) ? src[2] : 0.f;
                val.w = (kb + col + 3 < K) ? src[3] : 0.f;
            }
            *(v4f*)&As[row * SA + col] = val;
        }
        // ---- stage W chunk: rows n0..n0+63 (zero-padded past N), cols kb..kb+KC ----
#pragma unroll
        for (int it = 0; it < (64 * KC) / (128 * 4); ++it) {   // 8 iters
            const int f   = tid * 4 + it * 512;
            const int row = f >> 6;
            const int col = f & (KC - 1);
            const bool rv = (n0 + row < N);
            const float* src = W + (size_t)(rv ? (n0 + row) : 0) * K + kb + col;
            v4f val;
            if (rv && k_aligned && (kb + col + 4 <= K)) {
                val = *(const v4f*)src;
            } else {
                val.x = (rv && kb + col + 0 < K) ? src[0] : 0.f;
                val.y = (rv && kb + col + 1 < K) ? src[1] : 0.f;
                val.z = (rv && kb + col + 2 < K) ? src[2] : 0.f;
                val.w = (rv && kb + col + 3 < K) ? src[3] : 0.f;
            }
            *(v4f*)&Ws[row * SA + col] = val;
        }
        __syncthreads();

        // prefetch next chunk's cache lines while this chunk computes
        if (kb + KC < K) {
            __builtin_prefetch(A + (size_t)(m0b + tid) * K + kb + KC, 0, 1);
            if (tid < 64 && n0 + tid < N)
                __builtin_prefetch(W + (size_t)(n0 + tid) * K + kb + KC, 0, 1);
        }

        // ---- WMMA over this chunk (zero padding keeps loop guard-free) ----
        const int rem = K - kb;
        const int kc  = (rem > KC) ? KC : ((rem + 3) & ~3);
        for (int kk = 0; kk < kc; kk += 4) {
            const int c0 = kk + 2 * half;
            v2f a0, a1, b[4];
            {
                const float* p0 = &As[(wave * 32 + l16) * SA + c0];
                const float* p1 = &As[(wave * 32 + 16 + l16) * SA + c0];
                a0.x = p0[0]; a0.y = p0[1];
                a1.x = p1[0]; a1.y = p1[1];
            }
#pragma unroll
            for (int u = 0; u < 4; ++u) {
                const float* pw = &Ws[(u * 16 + l16) * SA + c0];
                b[u].x = pw[0]; b[u].y = pw[1];
            }
#pragma unroll
            for (int u = 0; u < 4; ++u) {
                acc[0][u] = __builtin_amdgcn_wmma_f32_16x16x4_f32(
                    false, a0, false, b[u], (short)0, acc[0][u], false, false);
                acc[1][u] = __builtin_amdgcn_wmma_f32_16x16x4_f32(
                    false, a1, false, b[u], (short)0, acc[1][u], false, false);
            }
        }
        __syncthreads();
    }

    // Precompute remapped output rows: r = m0 + t*16 + half*8 + i
    size_t orow[2][8];
#pragma unroll
    for (int t = 0; t < 2; ++t)
#pragma unroll
        for (int i = 0; i < 8; ++i) {
            int r = m0 + t * 16 + half * 8 + i;
            size_t mapped = (size_t)(r / rowsPerB) * outStrideB + outOff + (r % rowsPerB);
            orow[t][i] = mapped * (size_t)N;
        }

#pragma unroll
    for (int u = 0; u < 4; ++u) {
        const int n = n0 + u * 16 + l16;
        if (n >= N) continue;
        float bv = bias ? bias[n] : 0.f;
        if (bias2) bv += bias2[n];
#pragma unroll
        for (int t = 0; t < 2; ++t) {
            const float* ap = (const float*)&acc[t][u];
#pragma unroll
            for (int i = 0; i < 8; ++i) {
                float v = ap[i] + bv;
                if (relu) v = fmaxf(v, 0.f);
                C[orow[t][i] + n] = v;
            }
        }
    }
}

// ---------------------------------------------------------------------------
// Streaming attention with online softmax. One block per (b,h); one wave per
// query (wave32: lane holds head-dims 2*lane, 2*lane+1 of dh=64).
// ---------------------------------------------------------------------------
__global__ __launch_bounds__(256)
void attention_kernel(const float* __restrict__ Q, int qStride,
                      const float* __restrict__ Kp, int kStride,
                      const float* __restrict__ Vp, int vStride,
                      int Lk, float* __restrict__ Out, int outStride, int causal)
{
    const int lane = threadIdx.x & 31;
    const int wave = threadIdx.x >> 5;
    const int b = blockIdx.x / Hh;
    const int h = blockIdx.x % Hh;
    const float scale = 0.125f;  // 1/sqrt(64)

    for (int qi = wave; qi < Sv; qi += 8) {
        const float* qrow = Q + (size_t)(b * Sv + qi) * qStride + h * 64 + 2 * lane;
        const float qx = qrow[0], qy = qrow[1];
        float m = -1e30f, lsum = 0.f, ax = 0.f, ay = 0.f;
        const int kmax = causal ? (qi + 1) : Lk;
        const float* kbase = Kp + (size_t)b * Lk * kStride + h * 64 + 2 * lane;
        const float* vbase = Vp + (size_t)b * Lk * vStride + h * 64 + 2 * lane;
        for (int j = 0; j < kmax; ++j) {
            const float* kr = kbase + (size_t)j * kStride;
            float d = qx * kr[0] + qy * kr[1];
            d += __shfl_xor(d, 16, 32);
            d += __shfl_xor(d, 8, 32);
            d += __shfl_xor(d, 4, 32);
            d += __shfl_xor(d, 2, 32);
            d += __shfl_xor(d, 1, 32);
            const float s = d * scale;
            const float nm = fmaxf(m, s);
            const float corr = __expf(m - nm);
            const float p = __expf(s - nm);
            const float* vr = vbase + (size_t)j * vStride;
            ax = ax * corr + p * vr[0];
            ay = ay * corr + p * vr[1];
            lsum = lsum * corr + p;
            m = nm;
        }
        const float inv = 1.f / lsum;
        float* orow = Out + (size_t)(b * Sv + qi) * outStride + h * 64 + 2 * lane;
        orow[0] = ax * inv;
        orow[1] = ay * inv;
    }
}

// ---------------------------------------------------------------------------
// Out = LayerNorm(X + Y) * w + b  (one wave per 512-wide row; may alias X)
// ---------------------------------------------------------------------------
__global__ __launch_bounds__(256)
void add_ln_kernel(const float* __restrict__ X, const float* __restrict__ Y,
                   const float* __restrict__ w, const float* __restrict__ bb,
                   float* __restrict__ Out, int rows)
{
    const int row = blockIdx.x * 8 + (threadIdx.x >> 5);
    if (row >= rows) return;
    const int lane = threadIdx.x & 31;
    const float* xr = X + (size_t)row * Dm;
    const float* yr = Y + (size_t)row * Dm;
    float v[16];
    float s = 0.f;
#pragma unroll
    for (int i = 0; i < 16; ++i) {
        int c = lane + i * 32;
        v[i] = xr[c] + yr[c];
        s += v[i];
    }
    s += __shfl_xor(s, 16, 32); s += __shfl_xor(s, 8, 32);
    s += __shfl_xor(s, 4, 32);  s += __shfl_xor(s, 2, 32);
    s += __shfl_xor(s, 1, 32);
    const float mean = s * (1.f / Dm);
    float vs = 0.f;
#pragma unroll
    for (int i = 0; i < 16; ++i) { float d = v[i] - mean; vs += d * d; }
    vs += __shfl_xor(vs, 16, 32); vs += __shfl_xor(vs, 8, 32);
    vs += __shfl_xor(vs, 4, 32);  vs += __shfl_xor(vs, 2, 32);
    vs += __shfl_xor(vs, 1, 32);
    const float rstd = rsqrtf(vs * (1.f / Dm) + 1e-5f);
    float* orow = Out + (size_t)row * Dm;
#pragma unroll
    for (int i = 0; i < 16; ++i) {
        int c = lane + i * 32;
        orow[c] = (v[i] - mean) * rstd * w[c] + bb[c];
    }
}

// x[b,s,:] += pos[s,:] + enc[b,:]
__global__ void add_pos_enc_kernel(float* __restrict__ x, const float* __restrict__ pos,
                                   const float* __restrict__ enc)
{
    const int idx = blockIdx.x * 256 + threadIdx.x;
    const int total = Bv * Sv * Dm;
    if (idx >= total) return;
    const int dcol = idx % Dm;
    const int rs = idx / Dm;
    const int s = rs % Sv;
    const int b = rs / Sv;
    x[idx] += pos[s * Dm + dcol] + enc[b * Dm + dcol];
}

// dec[b,0,:] = start_token; dec[b,s,:] = target[b,s-1,:]
__global__ void build_decin_kernel(const float* __restrict__ start,
                                   const float* __restrict__ target,
                                   float* __restrict__ dec)
{
    const int idx = blockIdx.x * 256 + threadIdx.x;
    const int total = Bv * Sv * Avn;
    if (idx >= total) return;
    const int a = idx % Avn;
    const int rs = idx / Avn;
    const int s = rs % Sv;
    const int b = rs / Sv;
    dec[idx] = (s == 0) ? start[a] : target[((size_t)b * Sv + (s - 1)) * Avn + a];
}

// ---------------------------------------------------------------------------
extern "C" void kernel_launch(void* const* d_in, const int* in_sizes, int n_in,
                              void* d_out, int out_size, void* d_ws, size_t ws_size,
                              hipStream_t stream)
{
    (void)in_sizes; (void)n_in; (void)out_size; (void)ws_size;
    const float* state     = (const float*)d_in[0];
    const float* seg_f     = (const float*)d_in[1];
    const float* det_f     = (const float*)d_in[2];
    const float* vis_f     = (const float*)d_in[3];
    const float* pt_f      = (const float*)d_in[4];
    const float* target    = (const float*)d_in[5];
    const float* W_act     = (const float*)d_in[6];
    const float* b_act     = (const float*)d_in[7];
    const float* pos_embed = (const float*)d_in[8];
    const float* state_pos = (const float*)d_in[9];
    const float* W_state   = (const float*)d_in[10];
    const float* b_state   = (const float*)d_in[11];
    const float* W_seg     = (const float*)d_in[12];
    const float* b_seg     = (const float*)d_in[13];
    const float* W_det     = (const float*)d_in[14];
    const float* b_det     = (const float*)d_in[15];
    const float* W_vis     = (const float*)d_in[16];
    const float* b_vis     = (const float*)d_in[17];
    const float* W_pt      = (const float*)d_in[18];
    const float* b_pt      = (const float*)d_in[19];
    const float* start_tok = (const float*)d_in[20];
    const float* Wqkv_s    = (const float*)d_in[21];
    const float* bqkv_s    = (const float*)d_in[22];
    const float* Wo_s      = (const float*)d_in[23];
    const float* bo_s      = (const float*)d_in[24];
    const float* Wqkv_c    = (const float*)d_in[25];
    const float* bqkv_c    = (const float*)d_in[26];
    const float* Wo_c      = (const float*)d_in[27];
    const float* bo_c      = (const float*)d_in[28];
    const float* W1        = (const float*)d_in[29];
    const float* b1        = (const float*)d_in[30];
    const float* W2        = (const float*)d_in[31];
    const float* b2        = (const float*)d_in[32];
    const float* ln1w      = (const float*)d_in[33];
    const float* ln1b      = (const float*)d_in[34];
    const float* ln2w      = (const float*)d_in[35];
    const float* ln2b      = (const float*)d_in[36];
    const float* ln3w      = (const float*)d_in[37];
    const float* ln3b      = (const float*)d_in[38];
    const float* Wh1       = (const float*)d_in[39];
    const float* bh1       = (const float*)d_in[40];
    const float* Wh2       = (const float*)d_in[41];
    const float* bh2       = (const float*)d_in[42];
    const float* Wh3       = (const float*)d_in[43];
    const float* bh3       = (const float*)d_in[44];
    float* out = (float*)d_out;

    // Workspace layout (floats)
    float* ws = (float*)d_ws;
    size_t off = 0;
    const size_t memN = (size_t)Bv * NMEM * Dm;
    float* mem  = ws + off; off += memN;
    float* kc   = ws + off; off += memN;
    float* vc   = ws + off; off += memN;
    float* x    = ws + off; off += (size_t)Bv * Sv * Dm;
    float* qkv  = ws + off; off += (size_t)Bv * Sv * 3 * Dm;
    float* attn = ws + off; off += (size_t)Bv * Sv * Dm;
    float* proj = ws + off; off += (size_t)Bv * Sv * Dm;
    float* qc   = ws + off; off += (size_t)Bv * Sv * Dm;
    float* ffn  = ws + off; off += (size_t)Bv * Sv * Fv;
    float* hb1  = ws + off; off += (size_t)Bv * Sv * 2 * Dm;
    float* hb2  = ws + off; off += (size_t)Bv * Sv * Dm;
    float* tst  = ws + off; off += (size_t)Bv * Dm;
    float* enc  = ws + off; off += (size_t)Bv * Dm;
    float* dec  = ws + off; off += (size_t)Bv * Sv * Avn;

    auto gemm = [&](const float* A, const float* W, const float* bias, const float* bias2,
                    float* C, int M, int N, int K,
                    int rowsPerB, int strideB, int outOff, int relu) {
        dim3 grid((N + 63) / 64, M / 128);
        gemm_wmma_f32<<<grid, 128, 0, stream>>>(A, W, bias, bias2, C, M, N, K,
                                                rowsPerB, strideB, outOff, relu);
    };

    const int rowsBS = Bv * Sv;  // 3840

    // 1) enc = (state @ W_act^T + b_act + state_pos) @ W_state^T + b_state
    gemm(state, W_act, b_act, state_pos, tst, Bv, Dm, Avn, Bv, 0, 0, 0);
    gemm(tst, W_state, b_state, nullptr, enc, Bv, Dm, Dm, Bv, 0, 0, 0);

    // 2) memory: four encoders write directly into concatenated mem[B,1480,D]
    gemm(seg_f, W_seg, b_seg, nullptr, mem, Bv * NSEG, Dm, Dm, NSEG, NMEM, 0, 0);
    gemm(det_f, W_det, b_det, nullptr, mem, Bv * NDET, Dm, Dm, NDET, NMEM, NSEG, 0);
    gemm(vis_f, W_vis, b_vis, nullptr, mem, Bv * NVIS, Dm, Dm, NVIS, NMEM, NSEG + NDET, 0);
    gemm(pt_f,  W_pt,  b_pt,  nullptr, mem, Bv * NPT,  Dm, Dm, NPT,  NMEM, NSEG + NDET + NVIS, 0);

    // 3) decoder input embedding: x = dec_in @ W_act^T + b_act + pos + enc
    {
        int total = Bv * Sv * Avn;
        build_decin_kernel<<<(total + 255) / 256, 256, 0, stream>>>(start_tok, target, dec);
    }
    gemm(dec, W_act, b_act, nullptr, x, rowsBS, Dm, Avn, rowsBS, 0, 0, 0);
    {
        int total = Bv * Sv * Dm;
        add_pos_enc_kernel<<<(total + 255) / 256, 256, 0, stream>>>(x, pos_embed, enc);
    }

    // 4) decoder layers
    for (int i = 0; i < Lw; ++i) {
        const float* Wqs = Wqkv_s + (size_t)i * 3 * Dm * Dm;
        const float* bqs = bqkv_s + (size_t)i * 3 * Dm;
        const float* Wos = Wo_s + (size_t)i * Dm * Dm;
        const float* bos = bo_s + (size_t)i * Dm;
        const float* Wqc = Wqkv_c + (size_t)i * 3 * Dm * Dm;
        const float* bqc = bqkv_c + (size_t)i * 3 * Dm;
        const float* Woc = Wo_c + (size_t)i * Dm * Dm;
        const float* boc = bo_c + (size_t)i * Dm;

        // self-attention (causal)
        gemm(x, Wqs, bqs, nullptr, qkv, rowsBS, 3 * Dm, Dm, rowsBS, 0, 0, 0);
        attention_kernel<<<Bv * Hh, 256, 0, stream>>>(
            qkv, 3 * Dm, qkv + Dm, 3 * Dm, qkv + 2 * Dm, 3 * Dm, Sv, attn, Dm, 1);
        gemm(attn, Wos, bos, nullptr, proj, rowsBS, Dm, Dm, rowsBS, 0, 0, 0);
        add_ln_kernel<<<(rowsBS + 7) / 8, 256, 0, stream>>>(
            x, proj, ln1w + i * Dm, ln1b + i * Dm, x, rowsBS);

        // cross-attention over mem (K/V projected once per layer)
        gemm(x, Wqc, bqc, nullptr, qc, rowsBS, Dm, Dm, rowsBS, 0, 0, 0);
        gemm(mem, Wqc + (size_t)Dm * Dm, bqc + Dm, nullptr, kc,
             Bv * NMEM, Dm, Dm, Bv * NMEM, 0, 0, 0);
        gemm(mem, Wqc + (size_t)2 * Dm * Dm, bqc + 2 * Dm, nullptr, vc,
             Bv * NMEM, Dm, Dm, Bv * NMEM, 0, 0, 0);
        attention_kernel<<<Bv * Hh, 256, 0, stream>>>(
            qc, Dm, kc, Dm, vc, Dm, NMEM, attn, Dm, 0);
        gemm(attn, Woc, boc, nullptr, proj, rowsBS, Dm, Dm, rowsBS, 0, 0, 0);
        add_ln_kernel<<<(rowsBS + 7) / 8, 256, 0, stream>>>(
            x, proj, ln2w + i * Dm, ln2b + i * Dm, x, rowsBS);

        // FFN
        gemm(x, W1 + (size_t)i * Fv * Dm, b1 + (size_t)i * Fv, nullptr, ffn,
             rowsBS, Fv, Dm, rowsBS, 0, 0, 1);
        gemm(ffn, W2 + (size_t)i * Dm * Fv, b2 + (size_t)i * Dm, nullptr, proj,
             rowsBS, Dm, Fv, rowsBS, 0, 0, 0);
        add_ln_kernel<<<(rowsBS + 7) / 8, 256, 0, stream>>>(
            x, proj, ln3w + i * Dm, ln3b + i * Dm, x, rowsBS);
    }

    // 5) output head: D -> 2D -> D -> A
    gemm(x, Wh1, bh1, nullptr, hb1, rowsBS, 2 * Dm, Dm, rowsBS, 0, 0, 1);
    gemm(hb1, Wh2, bh2, nullptr, hb2, rowsBS, Dm, 2 * Dm, rowsBS, 0, 0, 1);
    gemm(hb2, Wh3, bh3, nullptr, out, rowsBS, Avn, Dm, rowsBS, 0, 0, 0);
}